// OriginalSkewSymmetricMatrix_14963666059946
// MI455X (gfx1250) — compile-verified
//
#include <hip/hip_runtime.h>
#include <math.h>

// ---------------------------------------------------------------------------
// Problem constants (from reference)
// ---------------------------------------------------------------------------
#define N_MAT   256        // skew-symmetric matrix size
#define TRI     32640      // N*(N-1)/2 strict lower-triangle elements used
#define BATCH   1024
#define HID     1024
#define IN_DIM  256
#define N_PAD2  32768      // padded N for layer-2 GEMM (multiple of 256, < 32896)

// ---------------------------------------------------------------------------
// GEMM tiling
// ---------------------------------------------------------------------------
#define KC      32         // K chunk staged in LDS
#define LDSLD   (KC + 4)   // padded LDS row stride (36 floats: 16B-aligned rows, conflict-free)
#define BM      32         // workgroup M tile
#define BN      256        // workgroup N tile
// 8 waves: 2 (M) x 4 (N); each wave: 16(M) x 64(N) = 4 accumulators of 16x16

#define SMEM_FLOATS (2 * (BM + BN) * LDSLD)          // double-buffered A+W tiles
#define SMEM_BYTES  (SMEM_FLOATS * sizeof(float))    // 82,944 B (dynamic LDS; WGP has 320KB)

typedef __attribute__((ext_vector_type(2))) float v2f;
typedef __attribute__((ext_vector_type(8))) float v8f;

// Async DMA: 16 bytes global -> LDS, tracked by ASYNCcnt (CDNA5 §15.18, op 98).
__device__ __forceinline__ void async_copy16(unsigned lds_off, const float* gptr)
{
    asm volatile("global_load_async_to_lds_b128 %0, %1, off"
                 :: "v"(lds_off), "v"(gptr) : "memory");
}
__device__ __forceinline__ void wait_async0()
{
    asm volatile("s_wait_asynccnt 0x0" ::: "memory");
}

// C[m,n] = sum_k A[m,k] * W[n,k]   (A: [M x K] row-major, W: [N x K] row-major)
// MODE 0: Cout[m*ldc + n] = softplus(C + bias[n])
// MODE 1: scatter skew-symmetric: e=n; (i,j)=tril_index(e);
//         out[m,i,j] = C+bias[e]; out[m,j,i] = -(C+bias[e])
template <int MODE>
__global__ __launch_bounds__(256)
void gemm_xwT_kernel(const float* __restrict__ A,
                     const float* __restrict__ W,
                     const float* __restrict__ bias,
                     float* __restrict__ Cout,
                     int K, int ldc)
{
    extern __shared__ float smem[];
    float* sA = smem;                       // [2][BM][LDSLD]
    float* sW = smem + 2 * BM * LDSLD;      // [2][BN][LDSLD]

    const int tid  = threadIdx.x;
    const int lane = tid & 31;
    const int wave = tid >> 5;
    const int wm   = wave >> 2;     // 0..1 : wave M position
    const int wn   = wave & 3;      // 0..3 : wave N position
    const int l16  = lane & 15;
    const int lh   = lane >> 4;     // 0/1 : which K-half this lane holds

    const int nBase = blockIdx.x * BN;
    const int mBase = blockIdx.y * BM;

    // staging assignment: one float4 per thread for A, eight for W
    const int arow = tid >> 3;          // 0..31
    const int acol = (tid & 7) * 4;     // 0..28

    v8f acc[4] = {};

    // issue async copies of one K-chunk into LDS buffer `buf`
    auto issue_chunk = [&](int buf, int k0) {
        async_copy16((unsigned)(size_t)&sA[(buf * BM + arow) * LDSLD + acol],
                     &A[(size_t)(mBase + arow) * K + k0 + acol]);
        #pragma unroll
        for (int r8 = 0; r8 < 8; ++r8) {
            const int row = r8 * 32 + arow;
            async_copy16((unsigned)(size_t)&sW[(buf * BN + row) * LDSLD + acol],
                         &W[(size_t)(nBase + row) * K + k0 + acol]);
        }
    };

    // WMMA over one staged chunk, with 1-step fragment software pipeline
    auto compute_chunk = [&](int buf) {
        const float* ap = &sA[(buf * BM + wm * 16 + l16) * LDSLD + 2 * lh];
        const float* bp0 = &sW[(buf * BN + wn * 64 +  0 + l16) * LDSLD + 2 * lh];
        const float* bp1 = &sW[(buf * BN + wn * 64 + 16 + l16) * LDSLD + 2 * lh];
        const float* bp2 = &sW[(buf * BN + wn * 64 + 32 + l16) * LDSLD + 2 * lh];
        const float* bp3 = &sW[(buf * BN + wn * 64 + 48 + l16) * LDSLD + 2 * lh];
        const float* bp[4] = {bp0, bp1, bp2, bp3};

        v2f a0, a1;
        v2f b0[4], b1[4];
        a0.x = ap[0]; a0.y = ap[1];
        #pragma unroll
        for (int t = 0; t < 4; ++t) { b0[t].x = bp[t][0]; b0[t].y = bp[t][1]; }

        #pragma unroll
        for (int kk = 0; kk < KC; kk += 4) {
            if (kk + 4 < KC) {          // prefetch next k-step's fragments
                a1.x = ap[kk + 4]; a1.y = ap[kk + 5];
                #pragma unroll
                for (int t = 0; t < 4; ++t) {
                    b1[t].x = bp[t][kk + 4]; b1[t].y = bp[t][kk + 5];
                }
            }
            #pragma unroll
            for (int t = 0; t < 4; ++t) {
                acc[t] = __builtin_amdgcn_wmma_f32_16x16x4_f32(
                    false, a0, false, b0[t], (short)0, acc[t], false, false);
            }
            a0 = a1;
            #pragma unroll
            for (int t = 0; t < 4; ++t) b0[t] = b1[t];
        }
    };

    // ---------------- pipelined main loop ----------------
    issue_chunk(0, 0);
    wait_async0();
    __syncthreads();

    for (int k0 = 0; k0 < K; k0 += KC) {
        const int cur = (k0 / KC) & 1;
        if (k0 + KC < K) issue_chunk(cur ^ 1, k0 + KC);   // overlap DMA with WMMA
        compute_chunk(cur);
        wait_async0();        // own async copies landed
        __syncthreads();      // => all waves' copies landed, all reads of `cur` done
    }

    // ------------------------------------------------------------------
    // Epilogue. C/D layout: value (m = r + 8*lh, n = l16) in acc[t][r].
    // ------------------------------------------------------------------
    #pragma unroll
    for (int t = 0; t < 4; ++t) {
        const int n = nBase + wn * 64 + t * 16 + l16;

        if (MODE == 0) {
            const float bv = bias[n];
            #pragma unroll
            for (int r = 0; r < 8; ++r) {
                const int m = mBase + wm * 16 + r + 8 * lh;
                float v = acc[t][r] + bv;
                // numerically stable softplus
                float sp = fmaxf(v, 0.0f) + log1pf(expf(-fabsf(v)));
                Cout[(size_t)m * ldc + n] = sp;
            }
        } else {
            if (n < TRI) {
                const float bv = bias[n];
                // invert triangular index: n = i*(i-1)/2 + j, 0 <= j < i
                int i = (int)floorf((1.0f + sqrtf(1.0f + 8.0f * (float)n)) * 0.5f);
                while (i * (i - 1) / 2 > n) --i;
                while ((i + 1) * i / 2 <= n) ++i;
                const int j = n - i * (i - 1) / 2;
                #pragma unroll
                for (int r = 0; r < 8; ++r) {
                    const int m = mBase + wm * 16 + r + 8 * lh;
                    float v = acc[t][r] + bv;
                    float* base = Cout + (size_t)m * (N_MAT * N_MAT);
                    base[i * N_MAT + j] =  v;
                    base[j * N_MAT + i] = -v;
                }
            }
        }
    }
}

// zero the diagonal entries (only output elements not covered by the scatter)
__global__ void zero_diag_kernel(float* __restrict__ out)
{
    const int idx = blockIdx.x * blockDim.x + threadIdx.x;  // B * N_MAT threads
    const int b = idx >> 8;
    const int d = idx & 255;
    out[(size_t)b * (N_MAT * N_MAT) + d * (N_MAT + 1)] = 0.0f;
}

// ---------------------------------------------------------------------------
// Entry point
// ---------------------------------------------------------------------------
extern "C" void kernel_launch(void* const* d_in, const int* in_sizes, int n_in,
                              void* d_out, int out_size, void* d_ws, size_t ws_size,
                              hipStream_t stream)
{
    const float* x  = (const float*)d_in[0];   // [B, IN]
    const float* W1 = (const float*)d_in[1];   // [HID, IN]
    const float* b1 = (const float*)d_in[2];   // [HID]
    const float* W2 = (const float*)d_in[3];   // [NUM_EL, HID]
    const float* b2 = (const float*)d_in[4];   // [NUM_EL]
    float* out = (float*)d_out;                // [B, 256, 256]
    float* h   = (float*)d_ws;                 // [B, HID] = 4 MB scratch

    // diagonal zeros (rest of the output is fully covered by the scatter)
    zero_diag_kernel<<<(BATCH * N_MAT) / 256, 256, 0, stream>>>(out);

    // layer 1: h = softplus(x @ W1^T + b1)   M=1024, N=1024, K=256
    dim3 g1(HID / BN, BATCH / BM);
    gemm_xwT_kernel<0><<<g1, 256, SMEM_BYTES, stream>>>(x, W1, b1, h, IN_DIM, HID);

    // layer 2: scatter(h @ W2^T + b2)        M=1024, N=32768(pad), K=1024
    dim3 g2(N_PAD2 / BN, BATCH / BM);
    gemm_xwT_kernel<1><<<g2, 256, SMEM_BYTES, stream>>>(h, W2, b2, out, HID, 0);
}